// GraphAutoEncoder_6167573037730
// MI455X (gfx1250) — compile-verified
//
#include <hip/hip_runtime.h>

// GraphAutoEncoder for MI455X (gfx1250): fp32 WMMA GEMMs + L2-resident
// atomic scatter aggregation. See analysis: workload is L2/atomic bound,
// GEMM FLOPs negligible -> full-precision V_WMMA_F32_16X16X4_F32.

typedef __attribute__((ext_vector_type(2))) float v2f;
typedef __attribute__((ext_vector_type(8))) float v8f;

static __device__ __forceinline__ void atomic_add_f32(float* p, float v) {
  __hip_atomic_fetch_add(p, v, __ATOMIC_RELAXED, __HIP_MEMORY_SCOPE_AGENT);
}

// ---------------- degree / normalization ----------------
__global__ void deg_init_k(float* __restrict__ deg, int n) {
  int i = blockIdx.x * blockDim.x + threadIdx.x;
  if (i < n) deg[i] = 1.0f;  // self-loop
}

__global__ void deg_count_k(const long long* __restrict__ dst,
                            float* __restrict__ deg, int E) {
  int e = blockIdx.x * blockDim.x + threadIdx.x;
  if (e < E) atomic_add_f32(&deg[(int)dst[e]], 1.0f);
}

__global__ void dis_k(const float* __restrict__ deg, float* __restrict__ dis, int n) {
  int i = blockIdx.x * blockDim.x + threadIdx.x;
  if (i < n) dis[i] = rsqrtf(deg[i]);  // deg >= 1 guaranteed
}

// ---------------- WMMA GEMM: C[MxCOUT] = op(A)[MxK] @ W[KxCOUT] ----------------
// One wave per 16-row tile; COUT/16 accumulators reuse the A fragment.
// f32 16x16x4 per-lane layout (ISA 7.12.2):
//   A: lanes 0-15 -> K=k0,k0+1 in v0,v1 (M=lane); lanes 16-31 -> K=k0+2,k0+3
//   B: mirrored, N = lane&15
//   D: VGPR r -> row r (lanes 0-15) / row r+8 (lanes 16-31), col = lane&15
template <int K, int COUT, bool RELU_IN>
__global__ __launch_bounds__(256) void gemm_wmma_k(
    const float* __restrict__ A, const float* __restrict__ W,
    float* __restrict__ C, int M) {
  const int wave = blockIdx.x * (blockDim.x >> 5) + ((int)threadIdx.x >> 5);
  const int lane = (int)threadIdx.x & 31;
  const int row0 = wave * 16;
  if (row0 >= M) return;                 // wave-uniform exit: EXEC stays all-1 for WMMA
  const int half = lane >> 4;
  const int l16 = lane & 15;
  constexpr int NT = COUT / 16;

  v8f acc[NT] = {};

  const int arow = row0 + l16;
  const bool arow_ok = (arow < M);
  const float* __restrict__ aptr = A + (size_t)arow * K;

#pragma unroll
  for (int k0 = 0; k0 < K; k0 += 4) {
    const int ka = k0 + half * 2;
    float a0 = 0.f, a1 = 0.f;
    if (arow_ok) { a0 = aptr[ka]; a1 = aptr[ka + 1]; }
    if (RELU_IN) { a0 = fmaxf(a0, 0.f); a1 = fmaxf(a1, 0.f); }
    v2f av;
    av.x = a0; av.y = a1;
#pragma unroll
    for (int nt = 0; nt < NT; ++nt) {
      const float* __restrict__ wp = W + (size_t)ka * COUT + nt * 16 + l16;
      v2f bv;
      bv.x = wp[0];
      bv.y = wp[COUT];
      acc[nt] = __builtin_amdgcn_wmma_f32_16x16x4_f32(
          false, av, false, bv, (short)0, acc[nt], false, false);
    }
  }

#pragma unroll
  for (int nt = 0; nt < NT; ++nt) {
#pragma unroll
    for (int r = 0; r < 8; ++r) {
      const int row = row0 + half * 8 + r;
      if (row < M) C[(size_t)row * COUT + nt * 16 + l16] = acc[nt][r];
    }
  }
}

// ---------------- aggregation ----------------
// Self-loop contribution + bias (fully initializes out; no pre-zeroing).
// rep_div > 1: t has one row per graph (decode layer), row = i / rep_div.
__global__ void agg_init_k(const float* __restrict__ t, const float* __restrict__ dis,
                           const float* __restrict__ bias, float* __restrict__ out,
                           int n, int C, int rep_div) {
  int idx = blockIdx.x * blockDim.x + threadIdx.x;
  if (idx >= n * C) return;
  int i = idx / C;
  int c = idx - i * C;
  int srow = (rep_div == 1) ? i : (i / rep_div);
  float w = dis[i] * dis[i];
  out[idx] = w * t[(size_t)srow * C + c] + bias[c];
}

// Per (c4-chunk, edge): out[dst, c4*4..+3] += dis[s]*dis[d] * t[srow, ...]
// Edge-major so edge-index loads coalesce across lanes.
__global__ void agg_edges_k(const long long* __restrict__ src,
                            const long long* __restrict__ dst,
                            const float* __restrict__ t,
                            const float* __restrict__ dis,
                            float* __restrict__ out,
                            int E, int C, int rep_div) {
  int idx = blockIdx.x * blockDim.x + threadIdx.x;
  int C4 = C >> 2;
  if (idx >= E * C4) return;
  int e = idx % E;
  int c4 = idx / E;
  int s = (int)src[e];
  int d = (int)dst[e];
  float w = dis[s] * dis[d];
  int srow = (rep_div == 1) ? s : (s / rep_div);
  const float4 v = *reinterpret_cast<const float4*>(t + (size_t)srow * C + (size_t)c4 * 4);
  float* o = out + (size_t)d * C + (size_t)c4 * 4;
  atomic_add_f32(o + 0, w * v.x);
  atomic_add_f32(o + 1, w * v.y);
  atomic_add_f32(o + 2, w * v.z);
  atomic_add_f32(o + 3, w * v.w);
}

// ---------------- per-graph mean pool (L=32 channels) ----------------
__global__ __launch_bounds__(256) void pool_mean_k(const float* __restrict__ a2,
                                                   float* __restrict__ z, int npg) {
  __shared__ float sdata[256];
  int g = blockIdx.x;
  int tid = threadIdx.x;
  int c = tid & 31;     // channel
  int part = tid >> 5;  // 8 node-strides
  float s = 0.f;
  const float* base = a2 + (size_t)g * npg * 32;
  for (int i = part; i < npg; i += 8) s += base[(size_t)i * 32 + c];
  sdata[tid] = s;
  __syncthreads();
  if (part == 0) {
    float tot = s;
#pragma unroll
    for (int p = 1; p < 8; ++p) tot += sdata[p * 32 + c];
    z[g * 32 + c] = tot / (float)npg;
  }
}

// ---------------- launcher ----------------
extern "C" void kernel_launch(void* const* d_in, const int* in_sizes, int n_in,
                              void* d_out, int out_size, void* d_ws, size_t ws_size,
                              hipStream_t stream) {
  (void)n_in; (void)out_size; (void)ws_size;
  constexpr int IN = 16, H = 64, L = 32, G = 100;

  const float* x = (const float*)d_in[0];
  const long long* ei = (const long long*)d_in[1];  // int64 edge_index [2, E]
  // d_in[2] = batch (unused: equal-sized sorted graphs), d_in[3] = nodes_per_graph
  const float* W1 = (const float*)d_in[4];
  const float* b1 = (const float*)d_in[5];
  const float* W2 = (const float*)d_in[6];
  const float* b2 = (const float*)d_in[7];
  const float* W3 = (const float*)d_in[8];
  const float* b3 = (const float*)d_in[9];
  const float* W4 = (const float*)d_in[10];
  const float* b4 = (const float*)d_in[11];

  const int N = in_sizes[0] / IN;
  const int E = in_sizes[1] / 2;
  const int NPG = N / G;
  const long long* srcp = ei;
  const long long* dstp = ei + E;

  // workspace carve (floats): deg, dis, two NxH ping-pong buffers, z, zw
  float* ws = (float*)d_ws;
  float* deg = ws;   ws += N;
  float* dis = ws;   ws += N;
  float* bufA = ws;  ws += (size_t)N * H;
  float* bufB = ws;  ws += (size_t)N * H;
  float* z = ws;     ws += 128 * L;
  float* zw = ws;    ws += 128 * H;

  const int T = 256;
  auto blk = [](long n, int t) { return (int)((n + t - 1) / t); };
  auto gemm_grid = [](int M) { return ((M + 15) / 16 + 7) / 8; };  // 8 waves/block

  // normalization: deg (self-loop + in-degree), dis = rsqrt(deg)
  deg_init_k<<<blk(N, T), T, 0, stream>>>(deg, N);
  deg_count_k<<<blk(E, T), T, 0, stream>>>(dstp, deg, E);
  dis_k<<<blk(N, T), T, 0, stream>>>(deg, dis, N);

  // ---- encode layer 1: t1 = x @ W1 ; a1 = Agg(t1) + b1 (relu deferred) ----
  gemm_wmma_k<IN, H, false><<<gemm_grid(N), T, 0, stream>>>(x, W1, bufA, N);
  agg_init_k<<<blk((long)N * H, T), T, 0, stream>>>(bufA, dis, b1, bufB, N, H, 1);
  agg_edges_k<<<blk((long)E * (H / 4), T), T, 0, stream>>>(srcp, dstp, bufA, dis, bufB, E, H, 1);

  // ---- encode layer 2: t2 = relu(a1) @ W2 ; a2 = Agg(t2) + b2 ----
  gemm_wmma_k<H, L, true><<<gemm_grid(N), T, 0, stream>>>(bufB, W2, bufA, N);
  agg_init_k<<<blk((long)N * L, T), T, 0, stream>>>(bufA, dis, b2, bufB, N, L, 1);
  agg_edges_k<<<blk((long)E * (L / 4), T), T, 0, stream>>>(srcp, dstp, bufA, dis, bufB, E, L, 1);

  // ---- pool: z[g] = mean over graph ----
  pool_mean_k<<<G, T, 0, stream>>>(bufB, z, NPG);

  // ---- decode layer 3: zr rows repeat per graph -> compute zw = z @ W3 (G rows only) ----
  gemm_wmma_k<L, H, false><<<gemm_grid(G), T, 0, stream>>>(z, W3, zw, G);
  agg_init_k<<<blk((long)N * H, T), T, 0, stream>>>(zw, dis, b3, bufA, N, H, NPG);
  agg_edges_k<<<blk((long)E * (H / 4), T), T, 0, stream>>>(srcp, dstp, zw, dis, bufA, E, H, NPG);

  // ---- decode layer 4: t4 = relu(a3) @ W4 ; out = Agg(t4) + b4 ----
  gemm_wmma_k<H, IN, true><<<gemm_grid(N), T, 0, stream>>>(bufA, W4, bufB, N);
  float* out = (float*)d_out;
  agg_init_k<<<blk((long)N * IN, T), T, 0, stream>>>(bufB, dis, b4, out, N, IN, 1);
  agg_edges_k<<<blk((long)E * (IN / 4), T), T, 0, stream>>>(srcp, dstp, bufB, dis, out, E, IN, 1);
}